// Voxelizer3D_48610439856783
// MI455X (gfx1250) — compile-verified
//
#include <hip/hip_runtime.h>
#include <cstdint>
#include <cstddef>

#define V 48
#define FD 19
#define NVOX (V * V * V)          // 110592
#define NG (NVOX * FD)            // 2101248 floats
#define SLICE_F (V * FD)          // 912 floats per (z,y) slice block

typedef __attribute__((ext_vector_type(2))) float v2f;
typedef __attribute__((ext_vector_type(8))) float v8f;

// Monotonic uint key for float ordering (finite values): key increases with f.
__device__ __forceinline__ unsigned f2key(float f) {
    unsigned u = __float_as_uint(f);
    return (u & 0x80000000u) ? ~u : (u | 0x80000000u);
}
__device__ __forceinline__ float key2f(unsigned k) {
    unsigned u = (k & 0x80000000u) ? (k & 0x7fffffffu) : ~k;
    return __uint_as_float(u);
}
__device__ __forceinline__ unsigned umin_(unsigned a, unsigned b) { return a < b ? a : b; }
__device__ __forceinline__ unsigned umax_(unsigned a, unsigned b) { return a > b ? a : b; }

// ---------------------------------------------------------------------------
// Init: mm[0..2]=max-key (mins), mm[3..5]=0 (maxs); zero accumulation buffer.
// ---------------------------------------------------------------------------
__global__ void k_init(unsigned* __restrict__ mm, float* __restrict__ zbuf, int count) {
    if (blockIdx.x == 0 && threadIdx.x < 6)
        mm[threadIdx.x] = (threadIdx.x < 3) ? 0xFFFFFFFFu : 0u;
    int stride = gridDim.x * blockDim.x;
    for (int i = blockIdx.x * blockDim.x + threadIdx.x; i < count; i += stride)
        zbuf[i] = 0.0f;
}

// ---------------------------------------------------------------------------
// Masked min/max: per-thread accumulate, wave32 shfl_xor butterfly, one
// atomicMin/Max(u32) per wave.
// ---------------------------------------------------------------------------
__global__ void k_minmax(const float* __restrict__ xyz, int n, unsigned* __restrict__ mm) {
    unsigned kmn[3] = {0xFFFFFFFFu, 0xFFFFFFFFu, 0xFFFFFFFFu};
    unsigned kmx[3] = {0u, 0u, 0u};
    int stride = gridDim.x * blockDim.x;
    for (int i = blockIdx.x * blockDim.x + threadIdx.x; i < n; i += stride) {
        float a = xyz[3 * i + 0], b = xyz[3 * i + 1], c = xyz[3 * i + 2];
        if (a != 0.0f && b != 0.0f && c != 0.0f) {
            unsigned ka = f2key(a), kb = f2key(b), kc = f2key(c);
            kmn[0] = umin_(kmn[0], ka); kmx[0] = umax_(kmx[0], ka);
            kmn[1] = umin_(kmn[1], kb); kmx[1] = umax_(kmx[1], kb);
            kmn[2] = umin_(kmn[2], kc); kmx[2] = umax_(kmx[2], kc);
        }
    }
    for (int m = 16; m > 0; m >>= 1) {
#pragma unroll
        for (int d = 0; d < 3; ++d) {
            kmn[d] = umin_(kmn[d], (unsigned)__shfl_xor((int)kmn[d], m, 32));
            kmx[d] = umax_(kmx[d], (unsigned)__shfl_xor((int)kmx[d], m, 32));
        }
    }
    if ((threadIdx.x & 31) == 0) {
#pragma unroll
        for (int d = 0; d < 3; ++d) {
            atomicMin(&mm[d], kmn[d]);
            atomicMax(&mm[3 + d], kmx[d]);
        }
    }
}

__device__ __forceinline__ int vidx(float v, float mn, float mx) {
    return (int)((v - mn) / (mx - mn) * 47.0f);   // nonneg -> trunc == floor
}

// ---------------------------------------------------------------------------
// Phase A: one thread per atom; 19 global_atomic_add_f32 into bin[vox][f]
// (27x fewer atomics than a direct splat).
// ---------------------------------------------------------------------------
__global__ void k_bin_scatter(const float* __restrict__ xyz, const float* __restrict__ feat,
                              int n, const unsigned* __restrict__ mm, float* __restrict__ bin) {
    int i = blockIdx.x * blockDim.x + threadIdx.x;
    if (i >= n) return;
    float a = xyz[3 * i + 0], b = xyz[3 * i + 1], c = xyz[3 * i + 2];
    if (a == 0.0f || b == 0.0f || c == 0.0f) return;
    int cx = vidx(a, key2f(mm[0]), key2f(mm[3]));
    int cy = vidx(b, key2f(mm[1]), key2f(mm[4]));
    int cz = vidx(c, key2f(mm[2]), key2f(mm[5]));
    int vox = (cz * V + cy) * V + cx;
    const float* fp = feat + (size_t)i * FD;
    float* bp = bin + (size_t)vox * FD;
#pragma unroll
    for (int f = 0; f < FD; ++f) atomicAdd(&bp[f], fp[f]);
}

// ---------------------------------------------------------------------------
// Phase B pass 1 (x-direction 3-tap) as WMMA GEMM per (z,y) slice:
// T[48x19] = S[48x48] * B[48x19], S tridiagonal-of-ones.
// Block = 8 waves = 4 slices x 2 N-tiles; each wave does 3 M-tiles.
// B slice blocks staged to LDS via async-to-LDS (ASYNCcnt path); fragments
// come from ds_load + cndmask (no EXEC divergence, no loadcnt stall feeding
// the WMMAs). Band-aware K ranges: 16 WMMAs per wave.
// ---------------------------------------------------------------------------
#if __has_builtin(__builtin_amdgcn_wmma_f32_16x16x4_f32)
#define USE_WMMA_PASS_X 1

#if __has_builtin(__builtin_amdgcn_global_load_async_to_lds_b128)
#define HAS_ASYNC_LDS 1
typedef int i4 __attribute__((vector_size(16)));
typedef __attribute__((address_space(1))) i4 gi4;   // global int4*
typedef __attribute__((address_space(3))) i4 li4;   // LDS int4*
#else
#define HAS_ASYNC_LDS 0
#endif

__global__ void __launch_bounds__(256) k_pass_x_wmma(const float* __restrict__ bin,
                                                     float* __restrict__ tmp) {
    __shared__ float lds[4 * SLICE_F];              // 14592 B
    const int t = threadIdx.x;
    const int slice0 = blockIdx.x * 4;              // 4 slices per block
    const float* __restrict__ gsrc = bin + (size_t)slice0 * SLICE_F;

    // ---- stage 4 slices (3648 dwords = 912 x b128) into LDS ----
#if HAS_ASYNC_LDS
    for (int c = t; c < (4 * SLICE_F) / 4; c += 256) {
        __builtin_amdgcn_global_load_async_to_lds_b128(
            (gi4*)(gsrc + 4 * c), (li4*)&lds[4 * c], 0, 0);
    }
    asm volatile("s_wait_asynccnt 0x0" ::: "memory");
#else
    for (int c = t; c < 4 * SLICE_F; c += 256) lds[c] = gsrc[c];
#endif
    __syncthreads();

    const int wave  = t >> 5;        // 0..7
    const int lane  = t & 31;
    const int sInB  = wave >> 1;     // slice within block: 0..3
    const int ntile = wave & 1;      // feature tile: cols 16*ntile..+15
    const float* __restrict__ B = lds + sInB * SLICE_F;

    const int half  = lane >> 4;     // lanes 16..31 hold K+2 pair
    const int l     = lane & 15;
    const int fglob = ntile * 16 + l;
    const bool fok  = fglob < FD;
    const int  fcl  = fok ? fglob : (FD - 1);   // clamped col: always-valid ds_load

    float* __restrict__ T = tmp + (size_t)(slice0 + sInB) * SLICE_F;

#pragma unroll
    for (int m = 0; m < 3; ++m) {                // M-tiles: x rows 16m..16m+15
        const int mbase = m * 16;
        const int Mrow  = mbase + l;
        v8f acc = {};
#pragma unroll
        for (int j = 0; j < 6; ++j) {            // band-aware K steps
            const int k0 = mbase - 4 + 4 * j;
            if (k0 < 0 || k0 > V - 4) continue;  // compile-time pruned
            const int ka = k0 + half * 2;
            v2f a, b;
            a.x = (Mrow >= ka - 1 && Mrow <= ka + 1) ? 1.0f : 0.0f;  // S[Mrow][ka]
            a.y = (Mrow >= ka     && Mrow <= ka + 2) ? 1.0f : 0.0f;  // S[Mrow][ka+1]
            float bx = B[ka * FD + fcl];
            float by = B[(ka + 1) * FD + fcl];
            b.x = fok ? bx : 0.0f;               // cndmask, no branch
            b.y = fok ? by : 0.0f;
            acc = __builtin_amdgcn_wmma_f32_16x16x4_f32(false, a, false, b, (short)0, acc,
                                                        false, false);
        }
        if (fok) {
#pragma unroll
            for (int r = 0; r < 8; ++r) {
                const int M = half * 8 + r;      // C/D: VGPR r = M=r (lanes 0-15) / M=8+r
                T[(mbase + M) * FD + fglob] = acc[r];
            }
        }
    }
}
#else
#define USE_WMMA_PASS_X 0
__global__ void k_pass_x_scalar(const float* __restrict__ src, float* __restrict__ dst) {
    int stride = gridDim.x * blockDim.x;
    for (int idx = blockIdx.x * blockDim.x + threadIdx.x; idx < NG; idx += stride) {
        int x = (idx / FD) % V;
        float s = src[idx];
        if (x > 0)     s += src[idx - FD];
        if (x < V - 1) s += src[idx + FD];
        dst[idx] = s;
    }
}
#endif

// Pass 2: y-direction 3-tap (stride V*FD), tmp -> bin.
__global__ void k_pass_y(const float* __restrict__ src, float* __restrict__ dst) {
    int stride = gridDim.x * blockDim.x;
    for (int idx = blockIdx.x * blockDim.x + threadIdx.x; idx < NG; idx += stride) {
        int y = (idx / (FD * V)) % V;
        float s = src[idx];
        if (y > 0)     s += src[idx - FD * V];
        if (y < V - 1) s += src[idx + FD * V];
        dst[idx] = s;
    }
}

// Pass 3: z-direction 3-tap fused with [vox,f] -> [f,z,y,x] transpose.
__global__ void k_pass_zt(const float* __restrict__ src, float* __restrict__ dst) {
    int stride = gridDim.x * blockDim.x;
    for (int o = blockIdx.x * blockDim.x + threadIdx.x; o < NG; o += stride) {
        int x = o % V;
        int y = (o / V) % V;
        int z = (o / (V * V)) % V;
        int f = o / (V * V * V);
        int idx = ((z * V + y) * V + x) * FD + f;
        float s = src[idx];
        if (z > 0)     s += src[idx - V * V * FD];
        if (z < V - 1) s += src[idx + V * V * FD];
        dst[o] = s;
    }
}

// ---------------------------------------------------------------------------
// Fallback (workspace too small): direct 27x19 atomic splat into d_out.
// ---------------------------------------------------------------------------
__global__ void k_splat(const float* __restrict__ xyz, const float* __restrict__ feat,
                        int n, const unsigned* __restrict__ mm, float* __restrict__ out) {
    int i = blockIdx.x * blockDim.x + threadIdx.x;
    if (i >= n) return;
    float a = xyz[3 * i + 0], b = xyz[3 * i + 1], c = xyz[3 * i + 2];
    if (a == 0.0f || b == 0.0f || c == 0.0f) return;
    int cx = vidx(a, key2f(mm[0]), key2f(mm[3]));
    int cy = vidx(b, key2f(mm[1]), key2f(mm[4]));
    int cz = vidx(c, key2f(mm[2]), key2f(mm[5]));
    const float* fp = feat + (size_t)i * FD;
    for (int dz = -1; dz <= 1; ++dz) {
        int pz = cz + dz; if (pz < 0 || pz > V - 1) continue;
        for (int dy = -1; dy <= 1; ++dy) {
            int py = cy + dy; if (py < 0 || py > V - 1) continue;
            for (int dx = -1; dx <= 1; ++dx) {
                int px = cx + dx; if (px < 0 || px > V - 1) continue;
                int voxo = (pz * V + py) * V + px;
#pragma unroll
                for (int f = 0; f < FD; ++f)
                    atomicAdd(&out[(size_t)f * NVOX + voxo], fp[f]);
            }
        }
    }
}

extern "C" void kernel_launch(void* const* d_in, const int* in_sizes, int n_in,
                              void* d_out, int out_size, void* d_ws, size_t ws_size,
                              hipStream_t stream) {
    const float* xyz  = (const float*)d_in[0];
    const float* feat = (const float*)d_in[1];
    float* out = (float*)d_out;
    int n = in_sizes[0] / 3;

    unsigned* mm = (unsigned*)d_ws;  // 6 u32 min/max keys at ws[0..24)
    const size_t need = 256 + 2 * (size_t)NG * sizeof(float);
    dim3 blk(256);

    if (ws_size >= need) {
        float* bin = (float*)((char*)d_ws + 256);
        float* tmp = bin + NG;
        k_init<<<2048, blk, 0, stream>>>(mm, bin, NG);
        k_minmax<<<512, blk, 0, stream>>>(xyz, n, mm);
        k_bin_scatter<<<(n + 255) / 256, blk, 0, stream>>>(xyz, feat, n, mm, bin);
#if USE_WMMA_PASS_X
        // 2304 slices / 4 per block = 576 blocks of 256 threads (8 waves)
        k_pass_x_wmma<<<576, blk, 0, stream>>>(bin, tmp);
#else
        k_pass_x_scalar<<<2048, blk, 0, stream>>>(bin, tmp);
#endif
        k_pass_y<<<2048, blk, 0, stream>>>(tmp, bin);
        k_pass_zt<<<2048, blk, 0, stream>>>(bin, out);
    } else {
        k_init<<<2048, blk, 0, stream>>>(mm, out, NG);
        k_minmax<<<512, blk, 0, stream>>>(xyz, n, mm);
        k_splat<<<(n + 255) / 256, blk, 0, stream>>>(xyz, feat, n, mm, out);
    }
}